// NerfExperts_5669356832627
// MI455X (gfx1250) — compile-verified
//
#include <hip/hip_runtime.h>
#include <math.h>

// ---------------------------------------------------------------------------
// NeRF mixture-of-experts forward, CDNA5 (gfx1250).
// bucket samples by expert -> one 16-row WMMA tile per (expert,tile) block ->
// stream each expert's fp32 weights exactly once through
// V_WMMA_F32_16X16X4_F32 (fp32 in / fp32 acc; exact vs fp32 reference).
// Bandwidth-bound: ~232MB weights, ~10us floor at 23.3 TB/s. Layers are
// templated on (DIN,DOUT) so all weight loads use immediate offsets from a
// strength-reduced base pointer (no per-iteration address math / clamps).
// ---------------------------------------------------------------------------

typedef float v2f __attribute__((ext_vector_type(2)));
typedef float v8f __attribute__((ext_vector_type(8)));

#define E_        100
#define H_        256
#define HD_       128
#define B_        1024
#define EMB_XYZ_  39
#define EMB_DIR_  27
#define LDA       297     // odd row stride (floats) -> conflict-free LDS banks
#define MAX_TILES 64      // 64*16 = 1024 rows: covers worst-case skew

struct Params {
  const float* w[8]; const float* b[8];
  const float* wi;  const float* bi;
  const float* wa;  const float* ba;
  const float* wc0; const float* bc0;
  const float* wc1; const float* bc1;
  const float* points; const float* dirs;
  const int*   index;
  const int*   counts; const int* list;
  float*       out;
};

// --- phase 1: bucket samples by expert -------------------------------------
__global__ void zero_counts_kernel(int* counts) {
  if (threadIdx.x < E_) counts[threadIdx.x] = 0;
}

__global__ void bucket_kernel(const int* __restrict__ index,
                              int* __restrict__ counts,
                              int* __restrict__ list) {
  int b = blockIdx.x * blockDim.x + threadIdx.x;
  if (b < B_) {
    int e = index[b];
    int slot = atomicAdd(&counts[e], 1);
    list[e * B_ + slot] = b;
  }
}

// --- fp32 WMMA dense layer: y[16,DOUT] = act(x[16,DIN] * W[DIN,DOUT] + b) ---
// A (16x4 f32): lane holds row (lane&15), K pair {2*(lane>>4), +1}.
// B (4x16 f32): lane holds col (lane&15), same K pair.
// C/D (16x16 f32): VGPR v -> row v + 8*(lane>>4), col (lane&15).
template<int DIN, int DOUT, bool RELU>
__device__ __forceinline__ void layer_wmma(const float* __restrict__ W,
                                           const float* __restrict__ Bv,
                                           const float* __restrict__ xin,
                                           float* __restrict__ yout) {
  constexpr int  KITERS = (DIN + 3) / 4;
  constexpr bool TAIL   = (DIN % 4) != 0;      // last step needs row clamp
  constexpr int  KMAIN  = TAIL ? KITERS - 1 : KITERS;

  const int lane = threadIdx.x & 31;
  const int wave = threadIdx.x >> 5;
  const int half = lane >> 4;                  // 0 | 1
  const int lid  = lane & 15;
  const int klo  = half * 2;
  const float* xrow = xin + lid * LDA + klo;   // A fragment source (LDS)

  for (int nt = wave; nt * 16 < DOUT; nt += 8) {
    const int n = nt * 16 + lid;
    const float* pW = W + klo * DOUT + n;      // per-lane weight column base
    v8f acc = {0.f, 0.f, 0.f, 0.f, 0.f, 0.f, 0.f, 0.f};

#pragma unroll 8
    for (int kk = 0; kk < KMAIN; ++kk) {
      v2f a;  a[0]  = xrow[kk * 4];            a[1]  = xrow[kk * 4 + 1];
      v2f bm; bm[0] = pW[kk * 4 * DOUT];       bm[1] = pW[(kk * 4 + 1) * DOUT];
      acc = __builtin_amdgcn_wmma_f32_16x16x4_f32(
          false, a, false, bm, (short)0, acc, false, false);
    }
    if (TAIL) {
      // k0 = KMAIN*4 + klo; for half==1 row k0+1 == DIN -> clamp (A elem is 0)
      int k0  = KMAIN * 4 + klo;
      int k0c = k0     < DIN ? k0     : DIN - 1;
      int k1c = k0 + 1 < DIN ? k0 + 1 : DIN - 1;
      v2f a;  a[0]  = xrow[KMAIN * 4];         a[1]  = xrow[KMAIN * 4 + 1];
      v2f bm; bm[0] = W[k0c * DOUT + n];       bm[1] = W[k1c * DOUT + n];
      acc = __builtin_amdgcn_wmma_f32_16x16x4_f32(
          false, a, false, bm, (short)0, acc, false, false);
    }

    float bias = Bv[nt * 16 + lid];
#pragma unroll
    for (int v = 0; v < 8; ++v) {
      int row = v + half * 8;
      float val = acc[v] + bias;
      if (RELU) val = fmaxf(val, 0.f);
      yout[row * LDA + nt * 16 + lid] = val;
    }
  }
}

// --- phase 2: full MLP for one 16-sample tile of one expert -----------------
__global__ __launch_bounds__(256, 1)
void nerf_experts_kernel(Params p) {
  const int e = blockIdx.x;
  const int t = blockIdx.y;
  const int cnt = p.counts[e];
  if (t * 16 >= cnt) return;             // uniform per block

  __shared__ float bufA[16 * LDA];
  __shared__ float bufB[16 * LDA];
  __shared__ int   sid_s[16];
  __shared__ float pts_s[16 * 3];
  __shared__ float dir_s[16 * 3];
  __shared__ float red_s[16 * 16];
  __shared__ float alpha_s[16];

  const int tid = threadIdx.x;

  // zero both activation buffers (covers all K padding)
  for (int i = tid; i < 16 * LDA; i += 256) { bufA[i] = 0.f; bufB[i] = 0.f; }
  if (tid < 16) {
    int m = t * 16 + tid;
    sid_s[tid] = (m < cnt) ? p.list[e * B_ + m] : -1;
  }
  __syncthreads();
  if (tid < 48) {
    int r = tid / 3, c = tid % 3;
    int s = sid_s[r];
    pts_s[tid] = (s >= 0) ? p.points[s * 3 + c] : 0.f;
    dir_s[tid] = (s >= 0) ? p.dirs[s * 3 + c]   : 0.f;
  }
  __syncthreads();

  // xyz harmonic embedding: [sin(18), cos(18), xyz(3)] -> bufA cols 0..38
  // (layer-0 input) and bufB cols 256..294 (skip-layer concat lives in bufB)
  for (int i = tid; i < 16 * EMB_XYZ_; i += 256) {
    int r = i / EMB_XYZ_, j = i % EMB_XYZ_;
    float v;
    if (j < 18)      { int c = j / 6, f = j % 6;               v = sinf(pts_s[r*3+c] * (float)(1 << f)); }
    else if (j < 36) { int jj = j - 18; int c = jj/6, f = jj%6; v = cosf(pts_s[r*3+c] * (float)(1 << f)); }
    else             v = pts_s[r * 3 + (j - 36)];
    bufA[r * LDA + j]       = v;
    bufB[r * LDA + 256 + j] = v;
  }
  __syncthreads();

  // 8 trunk layers with relu; ping-pong buffers (features end in bufA)
  layer_wmma<EMB_XYZ_,      H_, true>(p.w[0] + (size_t)e * EMB_XYZ_ * H_,        p.b[0] + (size_t)e * H_, bufA, bufB); __syncthreads();
  layer_wmma<H_,            H_, true>(p.w[1] + (size_t)e * H_ * H_,              p.b[1] + (size_t)e * H_, bufB, bufA); __syncthreads();
  layer_wmma<H_,            H_, true>(p.w[2] + (size_t)e * H_ * H_,              p.b[2] + (size_t)e * H_, bufA, bufB); __syncthreads();
  layer_wmma<H_,            H_, true>(p.w[3] + (size_t)e * H_ * H_,              p.b[3] + (size_t)e * H_, bufB, bufA); __syncthreads();
  layer_wmma<H_,            H_, true>(p.w[4] + (size_t)e * H_ * H_,              p.b[4] + (size_t)e * H_, bufA, bufB); __syncthreads();
  layer_wmma<H_ + EMB_XYZ_, H_, true>(p.w[5] + (size_t)e * (H_ + EMB_XYZ_) * H_, p.b[5] + (size_t)e * H_, bufB, bufA); __syncthreads();
  layer_wmma<H_,            H_, true>(p.w[6] + (size_t)e * H_ * H_,              p.b[6] + (size_t)e * H_, bufA, bufB); __syncthreads();
  layer_wmma<H_,            H_, true>(p.w[7] + (size_t)e * H_ * H_,              p.b[7] + (size_t)e * H_, bufB, bufA); __syncthreads();
  // features[16,256] now in bufA

  // alpha = features . wa + ba (no activation): tree reduction
  {
    int s = tid >> 4, ch = tid & 15;
    const float* wa_e = p.wa + (size_t)e * H_;
    float partial = 0.f;
    for (int k = ch * 16; k < ch * 16 + 16; ++k)
      partial += bufA[s * LDA + k] * wa_e[k];
    red_s[tid] = partial;
  }
  __syncthreads();
  if (tid < 16) {
    float a = 0.f;
    for (int j = 0; j < 16; ++j) a += red_s[tid * 16 + j];
    alpha_s[tid] = a + p.ba[e];
  }
  __syncthreads();

  // inter = features . wi + bi (no relu) -> bufB cols 0..255
  layer_wmma<H_, H_, false>(p.wi + (size_t)e * H_ * H_, p.bi + (size_t)e * H_,
                            bufA, bufB);
  __syncthreads();

  // direction harmonic embedding -> bufB cols 256..282, zero 283..287 pad
  for (int i = tid; i < 16 * 32; i += 256) {
    int r = i >> 5, j = i & 31;
    float v = 0.f;
    if (j < 12)      { int c = j >> 2, f = j & 3;               v = sinf(dir_s[r*3+c] * (float)(1 << f)); }
    else if (j < 24) { int jj = j - 12; int c = jj>>2, f = jj&3; v = cosf(dir_s[r*3+c] * (float)(1 << f)); }
    else if (j < 27) v = dir_s[r * 3 + (j - 24)];
    bufB[r * LDA + 256 + j] = v;
  }
  __syncthreads();

  // c = relu([inter, dir_emb] . wc0 + bc0) -> bufA cols 0..127
  layer_wmma<H_ + EMB_DIR_, HD_, true>(p.wc0 + (size_t)e * (H_ + EMB_DIR_) * HD_,
                                       p.bc0 + (size_t)e * HD_, bufB, bufA);
  __syncthreads();

  // color = sigmoid(c . wc1 + bc1); emit [alpha, rgb]
  if (tid < 48) {
    int s = tid / 3, o = tid % 3;
    const float* wc1_e = p.wc1 + (size_t)e * HD_ * 3;
    float acc = p.bc1[e * 3 + o];
    for (int k = 0; k < HD_; ++k)
      acc += bufA[s * LDA + k] * wc1_e[k * 3 + o];
    float col = 1.f / (1.f + expf(-acc));
    int sid = sid_s[s];
    if (sid >= 0) p.out[sid * 4 + 1 + o] = col;
  }
  if (tid >= 64 && tid < 80) {
    int s = tid - 64;
    int sid = sid_s[s];
    if (sid >= 0) p.out[sid * 4 + 0] = alpha_s[s];
  }
}

// ---------------------------------------------------------------------------
extern "C" void kernel_launch(void* const* d_in, const int* in_sizes, int n_in,
                              void* d_out, int out_size, void* d_ws, size_t ws_size,
                              hipStream_t stream) {
  (void)in_sizes; (void)n_in; (void)out_size; (void)ws_size;
  int* counts = (int*)d_ws;                       // E ints
  int* list   = (int*)((char*)d_ws + 512);        // E*B ints

  Params P;
  for (int i = 0; i < 8; ++i) {
    P.w[i] = (const float*)d_in[2 * i];
    P.b[i] = (const float*)d_in[2 * i + 1];
  }
  P.wi  = (const float*)d_in[16]; P.bi  = (const float*)d_in[17];
  P.wa  = (const float*)d_in[18]; P.ba  = (const float*)d_in[19];
  P.wc0 = (const float*)d_in[20]; P.bc0 = (const float*)d_in[21];
  P.wc1 = (const float*)d_in[22]; P.bc1 = (const float*)d_in[23];
  P.points = (const float*)d_in[24];
  P.dirs   = (const float*)d_in[25];
  P.index  = (const int*)d_in[26];
  P.counts = counts; P.list = list;
  P.out    = (float*)d_out;

  zero_counts_kernel<<<1, 128, 0, stream>>>(counts);
  bucket_kernel<<<(B_ + 255) / 256, 256, 0, stream>>>(P.index, counts, list);
  dim3 grid(E_, MAX_TILES);
  nerf_experts_kernel<<<grid, 256, 0, stream>>>(P);
}